// SA_Conv2d_2740189134891
// MI455X (gfx1250) — compile-verified
//
#include <hip/hip_runtime.h>
#include <hip/hip_bf16.h>

typedef __attribute__((ext_vector_type(2))) float v2f;
typedef __attribute__((ext_vector_type(8))) float v8f;
typedef __attribute__((ext_vector_type(4))) unsigned int u32x4;
typedef __attribute__((ext_vector_type(4))) int i32x4;
typedef __attribute__((ext_vector_type(8))) int i32x8;

#define HW   4096   // 64*64
#define HWP  1024   // 32*32
#define NB   8
#define CIN  512
#define CQK  64
#define CV   256
#define COUT 512
#define WROWS 32    // weight rows staged in LDS per block

// ---------------------------------------------------------------------------
// TDM: DMA a 2D f32 tile [rows x cols] (row stride = stride_elems) from global
// into LDS. D# per CDNA5 ISA:
//   group0: [0]=count=1, [1]=lds_addr, [2]=gaddr[31:0],
//           [3]=gaddr[56:32] | type=2<<30
//   group1: w0 = data_size(4B)=2<<16 | pad_enable<<20 | pad_interval(256dw)=7<<22
//           w1 = tensor_dim0[15:0]<<16          (bits 63:48)
//           w2 = tensor_dim1[15:0]<<16          (bits 95:80)
//           w3 = tile_dim0<<16                  (bits 127:112)
//           w4 = tile_dim1                      (bits 143:128)
//           w5 = tensor_dim0_stride[31:0]       (bits 191:160)
// pad: +1 DWORD after every 256 DWORDs stored (breaks the 512-dword row
// stride so A-operand lane reads don't all hit one LDS bank).
// ---------------------------------------------------------------------------
__device__ static inline void tdm_load_2d(const float* gsrc, float* ldst,
                                          int rows, int cols, int stride_elems,
                                          int pad_on) {
  unsigned long long ga = (unsigned long long)(uintptr_t)gsrc;
  u32x4 g0;
  g0.x = 1u;                                   // count=1, user descriptor
  g0.y = (unsigned)(uintptr_t)ldst;            // low 32 bits = LDS offset
  g0.z = (unsigned)(ga & 0xFFFFFFFFu);
  g0.w = (unsigned)((ga >> 32) & 0x1FFFFFFu) | (2u << 30);   // type=2 (image)
  i32x8 g1;
  g1[0] = (int)((2u << 16) | (pad_on ? ((1u << 20) | (7u << 22)) : 0u));
  g1[1] = (int)(((unsigned)cols & 0xFFFFu) << 16);   // tensor_dim0 lo
  g1[2] = (int)(((unsigned)rows & 0xFFFFu) << 16);   // tensor_dim1 lo
  g1[3] = (int)(((unsigned)cols & 0xFFFFu) << 16);   // tile_dim0
  g1[4] = rows;                                      // tile_dim1
  g1[5] = stride_elems;                              // tensor_dim0_stride lo
  g1[6] = 0;
  g1[7] = 0;
  i32x4 z4 = {0, 0, 0, 0};
#if __clang_major__ >= 23
  i32x8 z8 = {0, 0, 0, 0, 0, 0, 0, 0};
  __builtin_amdgcn_tensor_load_to_lds(g0, g1, z4, z4, z8, 0);
#else
  __builtin_amdgcn_tensor_load_to_lds(g0, g1, z4, z4, 0);
#endif
}

// index into the TDM-padded LDS image: +1 dword per 256 dwords
__device__ static inline int pad_idx(int i) { return i + (i >> 8); }

// ---------------------------------------------------------------------------
// 1x1-conv GEMM:  Y[b,o,n] = alpha*(sum_c W[o,c] X[b,c,n] + bias[o]) (+res)
// Block = 256 threads (8 waves): 8 n-subtiles of 16, each wave owns TWO
// o-tiles (ob+0..15, ob+16..31) sharing its B (x) loads -> 1 vmem + 2 ds
// per WMMA. Weight slab [32 x Cin] staged once via TDM into (padded) LDS.
// WMMA layouts per ISA 7.12.2: A lane=row M, VGPR0/1 = K+0/K+1 (half0),
// K+2/K+3 (half1); B lane=col N, VGPR0/1 = rows K+2h, K+2h+1; C VGPR r:
// rows r (lanes0-15) / r+8 (lanes16-31), col = lane%16.
// ---------------------------------------------------------------------------
__global__ __launch_bounds__(256) void conv1x1_wmma_kernel(
    const float* __restrict__ X, const float* __restrict__ W,
    const float* __restrict__ bias, const float* __restrict__ residual,
    const float* __restrict__ gamma_ptr, float* __restrict__ Y,
    int Cin, int Cout)
{
  __shared__ float wlds[WROWS * CIN + ((WROWS * CIN) >> 8)];  // 64KB + pad

  const int lane = threadIdx.x & 31;
  const int wave = threadIdx.x >> 5;
  const int lq   = lane & 15;
  const int half = lane >> 4;
  const int n0 = blockIdx.x * 128 + wave * 16;
  const int ob = blockIdx.y * WROWS;
  const int b  = blockIdx.z;
  const float* Xb = X + (size_t)b * Cin * HW;

  if (wave == 0) {
    tdm_load_2d(W + (size_t)ob * Cin, wlds, WROWS, Cin, Cin, /*pad=*/1);
    __builtin_amdgcn_s_wait_tensorcnt(0);
  }
  __syncthreads();

  v8f acc0 = {}, acc1 = {};
  #pragma unroll 4
  for (int k = 0; k < Cin; k += 4) {
    v2f bb;
    bb.x = Xb[(size_t)(k + 2 * half    ) * HW + n0 + lq];
    bb.y = Xb[(size_t)(k + 2 * half + 1) * HW + n0 + lq];
    const int i0 = lq * Cin + k + 2 * half;
    const int i1 = (16 + lq) * Cin + k + 2 * half;
    v2f a0;  a0.x = wlds[pad_idx(i0)];  a0.y = wlds[pad_idx(i0) + 1];
    v2f a1;  a1.x = wlds[pad_idx(i1)];  a1.y = wlds[pad_idx(i1) + 1];
    acc0 = __builtin_amdgcn_wmma_f32_16x16x4_f32(false, a0, false, bb,
                                                 (short)0, acc0, false, false);
    acc1 = __builtin_amdgcn_wmma_f32_16x16x4_f32(false, a1, false, bb,
                                                 (short)0, acc1, false, false);
  }

  const float alpha = gamma_ptr ? gamma_ptr[0] : 1.0f;
  const int n = n0 + lq;
  #pragma unroll
  for (int t = 0; t < 2; ++t) {
    #pragma unroll
    for (int r = 0; r < 8; ++r) {
      const int oc = ob + t * 16 + r + 8 * half;
      float vo = ((t ? acc1[r] : acc0[r]) + bias[oc]) * alpha;
      if (residual) vo += residual[((size_t)b * Cout + oc) * HW + n];
      Y[((size_t)b * Cout + oc) * HW + n] = vo;
    }
  }
}

// ---------------------------------------------------------------------------
// 2x2 max-pool [B,Cc,64,64] -> [B,Cc,32,32]; transpose_out=1 emits [B,mp,c].
// ---------------------------------------------------------------------------
__global__ __launch_bounds__(256) void maxpool_kernel(
    const float* __restrict__ src, float* __restrict__ dst,
    int Cc, int transpose_out, int total)
{
  int idx = blockIdx.x * 256 + threadIdx.x;
  if (idx >= total) return;
  const int per_b = Cc * HWP;
  const int b   = idx / per_b;
  const int rem = idx - b * per_b;
  const int c   = rem / HWP;
  const int mp  = rem - c * HWP;
  const int i = mp >> 5, j = mp & 31;
  const size_t base = ((size_t)b * Cc + c) * HW + (size_t)(2 * i) * 64 + 2 * j;
  float m0 = src[base],      m1 = src[base + 1];
  float m2 = src[base + 64], m3 = src[base + 65];
  float m = fmaxf(fmaxf(m0, m1), fmaxf(m2, m3));
  if (transpose_out) dst[((size_t)b * HWP + mp) * Cc + c] = m;
  else               dst[((size_t)b * Cc + c) * HWP + mp] = m;
}

// ---------------------------------------------------------------------------
// Fused flash-attention, one wave per (batch, 16-row n-tile):
//   s = f^T g  (K=64), streaming softmax over m, O += P @ vT.
// The 16KB vT tile (contiguous rows in [B,HWP,CV] layout) is TDM
// double-buffered into LDS: issue tile mt+1, s_wait_tensorcnt 1, compute
// tile mt from the front buffer (TDM ops of one wave complete in order).
// ---------------------------------------------------------------------------
__global__ __launch_bounds__(32) void attn_flash_kernel(
    const float* __restrict__ f, const float* __restrict__ g,
    const float* __restrict__ vT, float* __restrict__ o)
{
  const int lane = threadIdx.x & 31;
  const int lq   = lane & 15;
  const int half = lane >> 4;
  const int n0 = blockIdx.x * 16;
  const int b  = blockIdx.y;

  const float* fb = f  + (size_t)b * CQK * HW;
  const float* gb = g  + (size_t)b * CQK * HWP;
  const float* vb = vT + (size_t)b * HWP * CV;
  float*       ob = o  + (size_t)b * CV * HW;

  __shared__ float vlds[2][16 * CV];   // double-buffered vT tiles (2 x 16KB)
  __shared__ float pb[16 * 17];        // padded P-transpose staging

  // Preload f A-tile: 16 K-chunks of (16 n x 4 c).
  v2f fa[16];
  #pragma unroll
  for (int kc = 0; kc < 16; ++kc) {
    const int kb = 4 * kc + 2 * half;
    fa[kc].x = fb[(size_t)(kb    ) * HW + n0 + lq];
    fa[kc].y = fb[(size_t)(kb + 1) * HW + n0 + lq];
  }

  v8f acc[16];
  #pragma unroll
  for (int t = 0; t < 16; ++t) acc[t] = (v8f){};
  float mx[8], ls[8];
  #pragma unroll
  for (int r = 0; r < 8; ++r) { mx[r] = -3.0e38f; ls[r] = 0.0f; }

  // prime the async pipeline: vT tile 0 (16*256 contiguous floats)
  tdm_load_2d(vb, vlds[0], 1, 16 * CV, 16 * CV, /*pad=*/0);

  for (int mt = 0; mt < 64; ++mt) {
    const int m0 = mt * 16;

    // issue next tile, then wait for the current one (in-order completion)
    if (mt + 1 < 64) {
      tdm_load_2d(vb + (size_t)(mt + 1) * 16 * CV, vlds[(mt + 1) & 1],
                  1, 16 * CV, 16 * CV, 0);
      __builtin_amdgcn_s_wait_tensorcnt(1);
    } else {
      __builtin_amdgcn_s_wait_tensorcnt(0);
    }
    const float* vt = vlds[mt & 1];

    // ---- s tile: [16 n x 16 m], K=64 ----
    v8f s = {};
    #pragma unroll
    for (int kc = 0; kc < 16; ++kc) {
      const int kb = 4 * kc + 2 * half;
      v2f bb;
      bb.x = gb[(size_t)(kb    ) * HWP + m0 + lq];
      bb.y = gb[(size_t)(kb + 1) * HWP + m0 + lq];
      s = __builtin_amdgcn_wmma_f32_16x16x4_f32(false, fa[kc], false, bb,
                                                (short)0, s, false, false);
    }

    // ---- streaming softmax (row r+8*half of this lane's half) ----
    float al[8];
    #pragma unroll
    for (int r = 0; r < 8; ++r) {
      float rm = s[r];
      rm = fmaxf(rm, __shfl_xor(rm, 1));
      rm = fmaxf(rm, __shfl_xor(rm, 2));
      rm = fmaxf(rm, __shfl_xor(rm, 4));
      rm = fmaxf(rm, __shfl_xor(rm, 8));
      const float nm = fmaxf(mx[r], rm);
      al[r] = __expf(mx[r] - nm);
      mx[r] = nm;
      const float p = __expf(s[r] - nm);
      s[r] = p;
      float rs = p;
      rs += __shfl_xor(rs, 1);
      rs += __shfl_xor(rs, 2);
      rs += __shfl_xor(rs, 4);
      rs += __shfl_xor(rs, 8);
      ls[r] = ls[r] * al[r] + rs;
    }
    #pragma unroll
    for (int t = 0; t < 16; ++t)
      #pragma unroll
      for (int r = 0; r < 8; ++r) acc[t][r] *= al[r];

    // ---- transpose P (C-layout) -> A-layout via LDS ----
    #pragma unroll
    for (int r = 0; r < 8; ++r) pb[(r + 8 * half) * 17 + lq] = s[r];
    __syncthreads();
    v2f pa[4];
    #pragma unroll
    for (int kc = 0; kc < 4; ++kc) {
      const int kb = 4 * kc + 2 * half;
      pa[kc].x = pb[lq * 17 + kb];
      pa[kc].y = pb[lq * 17 + kb + 1];
    }
    __syncthreads();

    // ---- O += P (16n x 16m) @ vT tile (16m x 256c), from LDS ----
    #pragma unroll
    for (int t = 0; t < 16; ++t) {
      #pragma unroll
      for (int kc = 0; kc < 4; ++kc) {
        const int km = 4 * kc + 2 * half;
        v2f bb;
        bb.x = vt[(km    ) * CV + t * 16 + lq];
        bb.y = vt[(km + 1) * CV + t * 16 + lq];
        acc[t] = __builtin_amdgcn_wmma_f32_16x16x4_f32(false, pa[kc], false, bb,
                                                       (short)0, acc[t], false, false);
      }
    }
  }

  // ---- normalize and store o[c, n] ----
  #pragma unroll
  for (int t = 0; t < 16; ++t) {
    const int c = t * 16 + lq;
    #pragma unroll
    for (int r = 0; r < 8; ++r) {
      const int n = n0 + r + 8 * half;
      ob[(size_t)c * HW + n] = acc[t][r] / ls[r];
    }
  }
}

// ---------------------------------------------------------------------------
// Workspace layout (floats): f | g_full | h_full(=o later) | g | vT
// ---------------------------------------------------------------------------
#define OFF_F      ((size_t)0)
#define OFF_GFULL  (OFF_F     + (size_t)NB * CQK * HW)
#define OFF_HFULL  (OFF_GFULL + (size_t)NB * CQK * HW)
#define OFF_G      (OFF_HFULL + (size_t)NB * CV  * HW)
#define OFF_VT     (OFF_G     + (size_t)NB * CQK * HWP)
#define OFF_O      OFF_HFULL   // alias h_full after pooling

extern "C" void kernel_launch(void* const* d_in, const int* in_sizes, int n_in,
                              void* d_out, int out_size, void* d_ws, size_t ws_size,
                              hipStream_t stream) {
  const float* x     = (const float*)d_in[0];
  const float* Wf    = (const float*)d_in[1];
  const float* bf    = (const float*)d_in[2];
  const float* Wg    = (const float*)d_in[3];
  const float* bg    = (const float*)d_in[4];
  const float* Wh    = (const float*)d_in[5];
  const float* bh    = (const float*)d_in[6];
  const float* Wv    = (const float*)d_in[7];
  const float* bv    = (const float*)d_in[8];
  const float* gamma = (const float*)d_in[9];
  float* out = (float*)d_out;
  float* ws  = (float*)d_ws;

  float* f_buf  = ws + OFF_F;
  float* gfull  = ws + OFF_GFULL;
  float* hfull  = ws + OFF_HFULL;
  float* g_buf  = ws + OFF_G;
  float* vT_buf = ws + OFF_VT;
  float* o_buf  = ws + OFF_O;

  dim3 blk(256);
  // f = Wf @ x + bf                        [B, 64, 4096]
  conv1x1_wmma_kernel<<<dim3(HW / 128, CQK / WROWS, NB), blk, 0, stream>>>(
      x, Wf, bf, nullptr, nullptr, f_buf, CIN, CQK);
  // g_full = Wg @ x + bg                   [B, 64, 4096]
  conv1x1_wmma_kernel<<<dim3(HW / 128, CQK / WROWS, NB), blk, 0, stream>>>(
      x, Wg, bg, nullptr, nullptr, gfull, CIN, CQK);
  // h_full = Wh @ x + bh                   [B, 256, 4096]
  conv1x1_wmma_kernel<<<dim3(HW / 128, CV / WROWS, NB), blk, 0, stream>>>(
      x, Wh, bh, nullptr, nullptr, hfull, CIN, CV);

  // g = pool(g_full)  [B,64,1024];  vT = pool(h_full)^T  [B,1024,256]
  {
    int totg = NB * CQK * HWP;
    maxpool_kernel<<<(totg + 255) / 256, 256, 0, stream>>>(gfull, g_buf, CQK, 0, totg);
    int totv = NB * CV * HWP;
    maxpool_kernel<<<(totv + 255) / 256, 256, 0, stream>>>(hfull, vT_buf, CV, 1, totv);
  }

  // fused attention -> o [B, 256, 4096] (overwrites h_full region)
  attn_flash_kernel<<<dim3(HW / 16, NB), dim3(32), 0, stream>>>(
      f_buf, g_buf, vT_buf, o_buf);

  // out = gamma * (Wv @ o + bv) + x        [B, 512, 4096]
  conv1x1_wmma_kernel<<<dim3(HW / 128, COUT / WROWS, NB), blk, 0, stream>>>(
      o_buf, Wv, bv, x, gamma, out, CV, COUT);
}